// LCNN_38087769981042
// MI455X (gfx1250) — compile-verified
//
#include <hip/hip_runtime.h>

typedef float v2f __attribute__((ext_vector_type(2)));
typedef float v8f __attribute__((ext_vector_type(8)));
typedef unsigned int u32x4 __attribute__((ext_vector_type(4)));
typedef int i32x4 __attribute__((ext_vector_type(4)));
typedef int i32x8 __attribute__((ext_vector_type(8)));

constexpr int Bb = 64, Ll = 4096, Cc = 32, Kk = 8, Ff = 64;
constexpr int KC = Kk * Cc;           // 256 = reduction depth per location
constexpr int KCHUNK = 64;            // K-depth staged in LDS per chunk
constexpr int NCHUNK = KC / KCHUNK;   // 4
constexpr int STR = KCHUNK + 4;       // 68 floats: bank-friendly row stride

// Issue a TDM load of a 64-row x 64-float fp32 tile (row stride L*C elements)
// into LDS at lds_byte_off, with hardware row padding 64dw data + 4dw pad
// (reproduces STR=68 layout). D# built per CDNA5 ISA 8.3/8.4.
__device__ __forceinline__ void tdm_load_A(const float* gsrc, unsigned lds_byte_off) {
  unsigned long long ga = (unsigned long long)(uintptr_t)gsrc;
  u32x4 g0;
  g0[0] = 1u;                                           // count=1, user D#
  g0[1] = lds_byte_off;                                 // lds_addr
  g0[2] = (unsigned)(ga & 0xffffffffu);                 // global_addr[31:0]
  g0[3] = (unsigned)((ga >> 32) & 0x1ffffffu)           // global_addr[56:32]
        | (2u << 30);                                   // type=2 (image)
  i32x8 g1;
  g1[0] = (2 << 16)                                     // data_size = 4B
        | (1 << 20)                                     // pad_enable
        | (5 << 22)                                     // pad_interval: 64 dwords
        | (3 << 25);                                    // pad_amount: 4 dwords
  g1[1] = (KCHUNK & 0xffff) << 16;                      // tensor_dim0[15:0] = 64
  g1[2] = (Bb & 0xffff) << 16;                          // tensor_dim1[15:0] = 64
  g1[3] = (KCHUNK << 16);                               // tile_dim0 = 64
  g1[4] = Bb;                                           // tile_dim1 = 64, tile_dim2 = 0
  g1[5] = Ll * Cc;                                      // tensor_dim0_stride = 131072
  g1[6] = 0;
  g1[7] = 0;
  i32x4 gz4 = {0, 0, 0, 0};                             // groups 2/3: 2D tensor
  i32x8 gz8 = {0, 0, 0, 0, 0, 0, 0, 0};                 // trailing group: unused
  __builtin_amdgcn_tensor_load_to_lds(g0, g1, gz4, gz4, gz8, 0);
}

// One workgroup per location l. 8 waves; wave computes a 16x32 slab of the
// 64x64 (B x F) output via V_WMMA_F32_16X16X4_F32, K=256 over LDS-staged
// chunks. A tiles are DMA'd by the Tensor Data Mover, double-buffered.
__launch_bounds__(256)
__global__ void lcnn_wmma_f32(const float* __restrict__ x,
                              const float* __restrict__ w,
                              const float* __restrict__ bias,
                              float* __restrict__ out) {
  __shared__ float Alds[2][Bb * STR];  // double-buffered window rows [b][kc_local]
  __shared__ float Blds[Ff * STR];     // transposed weights [f][kc_local]

  const int l    = blockIdx.x;
  const int tid  = threadIdx.x;
  const int lane = tid & 31;
  const int wave = tid >> 5;
  const int half = lane >> 4;          // K-half (A/B frags) / M-half (C/D)
  const int lm   = lane & 15;

  const int tm  = wave & 3;            // M tile 0..3
  const int tn0 = (wave >> 2) << 1;    // first of two N tiles (0 or 2)

  const float* wl   = w + (size_t)l * (KC * Ff);
  const long  xbase = (long)(l - (Kk - 1)) * Cc;   // window start col in x row
  const int   minkc = (Kk - 1 - l) * Cc;           // kc below this is zero pad
  const bool  use_tdm = (l >= Kk - 1);             // interior: clean 2D tile

  v8f acc0 = {0.f,0.f,0.f,0.f,0.f,0.f,0.f,0.f};
  v8f acc1 = {0.f,0.f,0.f,0.f,0.f,0.f,0.f,0.f};

  const int ar = tid >> 2;             // A row staged by this thread (manual path)
  const int ac = (tid & 3) << 2;       // A col base (float4)

  // Kick off TDM for chunk 0 (one wave issues; TDM ignores EXEC).
  if (use_tdm && tid == 0) {
    tdm_load_A(x + xbase, (unsigned)(uintptr_t)(&Alds[0][0]));
  }

  for (int ch = 0; ch < NCHUNK; ++ch) {
    const int kc0 = ch * KCHUNK;
    float* Acur = &Alds[ch & 1][0];

    // ---- stage B chunk transposed: w[l][kc0+kcl][f] -> Blds[f][kcl] ----
    {
      const float* wc = wl + kc0 * Ff;
      #pragma unroll
      for (int j = 0; j < 4; ++j) {
        const int g   = (j * 256 + tid) * 4;   // coalesced float4 over 4096 floats
        float4 v      = *(const float4*)(wc + g);
        const int kcl = g >> 6;                // g / F
        const int f   = g & 63;                // g % F
        Blds[(f + 0) * STR + kcl] = v.x;       // banks kcl+{0,4,8,12}: conflict-free
        Blds[(f + 1) * STR + kcl] = v.y;
        Blds[(f + 2) * STR + kcl] = v.z;
        Blds[(f + 3) * STR + kcl] = v.w;
      }
    }

    if (use_tdm) {
      // Issue TDM for next chunk into the other buffer, then wait for the
      // current chunk's DMA (TENSORcnt is in-order per wave).
      if (ch + 1 < NCHUNK && tid == 0) {
        tdm_load_A(x + xbase + (ch + 1) * KCHUNK,
                   (unsigned)(uintptr_t)(&Alds[(ch + 1) & 1][0]));
      }
      if (ch + 1 < NCHUNK) __builtin_amdgcn_s_wait_tensorcnt(1);
      else                 __builtin_amdgcn_s_wait_tensorcnt(0);
    } else {
      // Edge blocks (l < K-1): manual staging with zero padding on the left.
      const float* xrow = x + (long)ar * (Ll * Cc) + xbase + kc0;
      #pragma unroll
      for (int j = 0; j < 4; ++j) {
        const int c = ac + j * 16;
        float4 v = make_float4(0.f, 0.f, 0.f, 0.f);
        if (kc0 + c >= minkc) v = *(const float4*)(xrow + c);
        *(float4*)(&Acur[ar * STR + c]) = v;
      }
    }

    // prefetch next streaming weight chunk (global_prefetch_b8)
    if (ch + 1 < NCHUNK) {
      __builtin_prefetch(wl + (kc0 + KCHUNK) * Ff + tid * 16, 0, 0);
    }
    __syncthreads();

    // ---- 16 K-steps of V_WMMA_F32_16X16X4_F32 ----
    // A frag (16x4): lane<16 -> K={kk,kk+1}; lane>=16 -> K={kk+2,kk+3}
    // B frag (4x16): same K split; N = lane%16 (Blds transposed -> b64 reads)
    const float* aRow  = &Acur[(tm * 16 + lm) * STR];
    const float* bRow0 = &Blds[(tn0 * 16 + lm) * STR];
    const float* bRow1 = &Blds[(tn0 * 16 + 16 + lm) * STR];
    #pragma unroll
    for (int kk = 0; kk < KCHUNK; kk += 4) {
      const int off = kk + half * 2;
      v2f a  = *(const v2f*)(aRow  + off);
      v2f b0 = *(const v2f*)(bRow0 + off);
      v2f b1 = *(const v2f*)(bRow1 + off);
      acc0 = __builtin_amdgcn_wmma_f32_16x16x4_f32(
          false, a, false, b0, (short)0, acc0, false, false);
      acc1 = __builtin_amdgcn_wmma_f32_16x16x4_f32(
          false, a, false, b1, (short)0, acc1, false, false);
    }
    __syncthreads();
  }

  // ---- epilogue: bias + store ----
  // C/D 16x16 f32 layout: VGPR v, lanes 0-15: M=v; lanes 16-31: M=v+8; N=lane%16
  const int n0 = tn0 * 16 + lm;
  const int n1 = n0 + 16;
  const float bs0 = bias[l * Ff + n0];
  const float bs1 = bias[l * Ff + n1];
  #pragma unroll
  for (int v = 0; v < 8; ++v) {
    const int m = tm * 16 + half * 8 + v;            // batch index
    float* o = out + ((long)m * Ll + l) * Ff;
    o[n0] = acc0[v] + bs0;
    o[n1] = acc1[v] + bs1;
  }
}

extern "C" void kernel_launch(void* const* d_in, const int* in_sizes, int n_in,
                              void* d_out, int out_size, void* d_ws, size_t ws_size,
                              hipStream_t stream) {
  const float* x    = (const float*)d_in[0];   // (B, L, C) fp32
  const float* w    = (const float*)d_in[1];   // (L, K, C, F) fp32
  const float* bias = (const float*)d_in[2];   // (L, F) fp32
  float* out = (float*)d_out;                  // (B, L, F) fp32

  dim3 grid(Ll);
  dim3 block(256);
  lcnn_wmma_f32<<<grid, block, 0, stream>>>(x, w, bias, out);
}